// iSHT_60498909331755
// MI455X (gfx1250) — compile-verified
//
#include <hip/hip_runtime.h>

#define NSIDE 64
#define NLAT  255
#define NLON  256
#define MMAX  129
#define LMAX  255
#define BC    512
#define NPIX  49152
#define XSTRIDE (LMAX * MMAX)   // 32895, stride between bc rows of x
#define PSTRIDE (LMAX * NLAT)   // 65025, stride between m slabs of pct
#define WSTRIDE (NLAT * MMAX)   // 32895, stride between bc rows of xc

#define TD_ROWS 256             // staged l rows; row 255 zero-filled via TDM OOB
#define TD_COLS 16              // one 16-wide ring tile per descriptor

typedef float v2f __attribute__((ext_vector_type(2)));
typedef float v8f __attribute__((ext_vector_type(8)));
typedef unsigned int u32x4 __attribute__((ext_vector_type(4)));
typedef int i32x8 __attribute__((ext_vector_type(8)));
typedef int i32x4 __attribute__((ext_vector_type(4)));

#if __has_builtin(__builtin_amdgcn_tensor_load_to_lds)
#define HAVE_TDM 1
#else
#define HAVE_TDM 0
#endif

#if HAVE_TDM
// DMA a [TD_ROWS x TD_COLS] fp32 tile of a row-major [LMAX x NLAT] slab into
// LDS. valid_cols / LMAX bound the tensor so TDM zero-fills the k>=NLAT edge
// columns and the padded l==255 row (OOB reads return zero).
__device__ __forceinline__ void tdm_load_tile(unsigned lds_byte,
                                              const float* gptr,
                                              int valid_cols) {
  unsigned long long ga = (unsigned long long)(size_t)gptr;
  u32x4 g0;
  g0[0] = 1u;                                    // count=1, user descriptor
  g0[1] = lds_byte;                              // lds_addr (bytes)
  g0[2] = (unsigned)(ga & 0xFFFFFFFFu);          // global_addr[31:0]
  g0[3] = (unsigned)((ga >> 32) & 0x01FFFFFFu)   // global_addr[56:32]
          | (2u << 30);                          // type=2 ("image")
  i32x8 g1;
  g1[0] = (2 << 16);                             // data_size = 4 bytes
  g1[1] = (valid_cols & 0xFFFF) << 16;           // tensor_dim0[15:0]
  g1[2] = (LMAX & 0xFFFF) << 16;                 // dim0 hi=0 | tensor_dim1 lo
  g1[3] = (TD_COLS << 16);                       // dim1 hi=0 | tile_dim0=16
  g1[4] = TD_ROWS;                               // tile_dim1=256 | tile_dim2=0
  g1[5] = NLAT;                                  // tensor_dim0_stride[31:0]
  g1[6] = 0;                                     // stride hi | dim1_stride lo
  g1[7] = 0;
  i32x4 z4 = {0, 0, 0, 0};
  i32x8 z8 = {0, 0, 0, 0, 0, 0, 0, 0};
  // 6-arg form (this toolchain): (g0, g1, g2, g3, g4, cpol)
  __builtin_amdgcn_tensor_load_to_lds(g0, g1, z4, z4, z8, 0);
}
#endif

// ---------------------------------------------------------------------------
// Kernel A: Legendre synthesis. For each m: [512 x 255] x [255 x 255] GEMM,
// real+imag with shared B fragments, fp32 WMMA 16x16x4.
// Block = 128 threads (4 waves): wave w owns bc row-tile rowQ*4+w; all waves
// share two TDM-staged 16-wide pct tiles in LDS (conflict-free stride-16).
// Grid: (colPairs=8, rowQuads=8, m=129).
// ---------------------------------------------------------------------------
__global__ __launch_bounds__(128) void legendre_wmma_kernel(
    const float* __restrict__ xr, const float* __restrict__ xi,
    const float* __restrict__ pct,
    float* __restrict__ wre, float* __restrict__ wim)
{
  __shared__ float Bs[2][TD_ROWS * TD_COLS];     // 2 x 16 KB

  const int m    = blockIdx.z;                   // 0..128
  const int rowQ = blockIdx.y;                   // 0..7
  const int colP = blockIdx.x;                   // 0..7
  const int tid  = threadIdx.x;
  const int wave = tid >> 5;
  const int lane = tid & 31;
  const int half = lane >> 4;                    // 0 or 1
  const int lid  = lane & 15;

  const int rowT = rowQ * 4 + wave;              // 0..31
  const int row  = rowT * 16 + lid;              // bc index (< 512)
  const int c0   = colP * 32;                    // tile0 cols c0.., tile1 c0+16..

#if HAVE_TDM
  if (tid < 32) {                                // wave 0 issues the DMAs
    const float* gp0 = pct + (size_t)m * PSTRIDE + c0;
    tdm_load_tile((unsigned)(size_t)&Bs[0][0], gp0,      NLAT - c0);
    tdm_load_tile((unsigned)(size_t)&Bs[1][0], gp0 + 16, NLAT - (c0 + 16));
    __builtin_amdgcn_s_wait_tensorcnt(0);
  }
#else
  for (int idx = tid; idx < 2 * TD_ROWS * TD_COLS; idx += 128) {
    const int t   = idx / (TD_ROWS * TD_COLS);
    const int rem = idx % (TD_ROWS * TD_COLS);
    const int l   = rem / TD_COLS;
    const int c   = c0 + t * 16 + (rem % TD_COLS);
    Bs[t][rem] = (l < LMAX && c < NLAT)
                     ? pct[(size_t)m * PSTRIDE + (size_t)l * NLAT + c] : 0.0f;
  }
#endif
  __syncthreads();

  const float* Ar = xr + (size_t)row * XSTRIDE + m;   // A[row][l] at +l*MMAX
  const float* Ai = xi + (size_t)row * XSTRIDE + m;

  v8f aR0 = {}, aI0 = {}, aR1 = {}, aI1 = {};

  int l = 0;
  for (; l + 4 <= LMAX; l += 4) {
    const int la = l + 2 * half;  // lanes 0-15: K=l,l+1 ; lanes 16-31: K=l+2,l+3
    v2f ar, ai, b0, b1;
    ar.x = Ar[(size_t)la * MMAX];
    ar.y = Ar[(size_t)(la + 1) * MMAX];
    ai.x = Ai[(size_t)la * MMAX];
    ai.y = Ai[(size_t)(la + 1) * MMAX];
    __builtin_prefetch(&Ar[(size_t)(la + 16) * MMAX], 0, 3);
    __builtin_prefetch(&Ai[(size_t)(la + 16) * MMAX], 0, 3);
    b0.x = Bs[0][la * TD_COLS + lid];
    b0.y = Bs[0][(la + 1) * TD_COLS + lid];
    b1.x = Bs[1][la * TD_COLS + lid];
    b1.y = Bs[1][(la + 1) * TD_COLS + lid];
    aR0 = __builtin_amdgcn_wmma_f32_16x16x4_f32(false, ar, false, b0,
                                                (short)0, aR0, false, false);
    aI0 = __builtin_amdgcn_wmma_f32_16x16x4_f32(false, ai, false, b0,
                                                (short)0, aI0, false, false);
    aR1 = __builtin_amdgcn_wmma_f32_16x16x4_f32(false, ar, false, b1,
                                                (short)0, aR1, false, false);
    aI1 = __builtin_amdgcn_wmma_f32_16x16x4_f32(false, ai, false, b1,
                                                (short)0, aI1, false, false);
  }

  { // K remainder l = 252..254; A row 255 masked, LDS B row 255 already zero
    const int la  = l + 2 * half;                      // 252 or 254
    const int la1 = (la + 1 < LMAX) ? (la + 1) : la;   // clamp A address
    const float km = (la + 1 < LMAX) ? 1.0f : 0.0f;    // mask A value
    v2f ar, ai, b0, b1;
    ar.x = Ar[(size_t)la * MMAX];
    ar.y = Ar[(size_t)la1 * MMAX] * km;
    ai.x = Ai[(size_t)la * MMAX];
    ai.y = Ai[(size_t)la1 * MMAX] * km;
    b0.x = Bs[0][la * TD_COLS + lid];
    b0.y = Bs[0][(la + 1) * TD_COLS + lid];
    b1.x = Bs[1][la * TD_COLS + lid];
    b1.y = Bs[1][(la + 1) * TD_COLS + lid];
    aR0 = __builtin_amdgcn_wmma_f32_16x16x4_f32(false, ar, false, b0,
                                                (short)0, aR0, false, false);
    aI0 = __builtin_amdgcn_wmma_f32_16x16x4_f32(false, ai, false, b0,
                                                (short)0, aI0, false, false);
    aR1 = __builtin_amdgcn_wmma_f32_16x16x4_f32(false, ar, false, b1,
                                                (short)0, aR1, false, false);
    aI1 = __builtin_amdgcn_wmma_f32_16x16x4_f32(false, ai, false, b1,
                                                (short)0, aI1, false, false);
  }

  // D layout (32-bit 16x16): lanes 0-15 -> rows M=j, lanes 16-31 -> rows M=8+j
  const int rbase = rowT * 16 + 8 * half;
  {
    const int col = c0 + lid;
    if (col < NLAT) {
#pragma unroll
      for (int j = 0; j < 8; ++j) {
        const size_t o = (size_t)(rbase + j) * WSTRIDE + (size_t)col * MMAX + m;
        wre[o] = aR0[j];
        wim[o] = aI0[j];
      }
    }
  }
  {
    const int col = c0 + 16 + lid;
    if (col < NLAT) {
#pragma unroll
      for (int j = 0; j < 8; ++j) {
        const size_t o = (size_t)(rbase + j) * WSTRIDE + (size_t)col * MMAX + m;
        wre[o] = aR1[j];
        wim[o] = aI1[j];
      }
    }
  }
}

// ---------------------------------------------------------------------------
// Kernel B: per-ring inverse real DFT with HEALPix ragged rings.
// One block per (ring, bc); coefficients staged in LDS; per-pixel complex
// rotation recurrence (1 sincos per thread, FMA-only loop).
// out[p] = (1/n) * sum_m w_m * (re_m*cos(m*alpha) - im_m*sin(m*alpha)),
// alpha = phi0 + 2*pi*p/n, w = 1 for m=0 and Nyquist, else 2.
// ---------------------------------------------------------------------------
__global__ __launch_bounds__(256) void healpix_irfft_kernel(
    const float* __restrict__ wre, const float* __restrict__ wim,
    float* __restrict__ out)
{
  __shared__ float lre[MMAX];
  __shared__ float lim[MMAX];

  const int r  = blockIdx.y;   // output ring index == xc ring index
  const int bc = blockIdx.z;
  const float PI = 3.14159265358979323846f;

  int nphi, start;
  float phi0;
  if (r < NSIDE - 1) {                       // north polar cap, k = r+1
    const int k = r + 1;
    nphi  = 4 * k;
    start = 2 * k * (k - 1);
    phi0  = PI / (float)nphi;
  } else if (r < 3 * NSIDE) {                // equatorial belt
    const int i = r - (NSIDE - 1);
    nphi  = NLON;
    start = 2 * NSIDE * (NSIDE - 1) + i * NLON;
    phi0  = ((i & 1) == 0) ? (PI / (float)NLON) : 0.0f;  // s=(i+1)%2
  } else {                                   // south polar cap, k = 4*nside-1-r
    const int k = 4 * NSIDE - 1 - r;
    nphi  = 4 * k;
    start = NPIX - 2 * k * (k + 1);
    phi0  = PI / (float)nphi;
  }
  const int nm = nphi / 2 + 1;

  const float* pre = wre + ((size_t)bc * NLAT + r) * MMAX;
  const float* pim = wim + ((size_t)bc * NLAT + r) * MMAX;
  for (int t = threadIdx.x; t < nm; t += blockDim.x) {
    lre[t] = pre[t];
    lim[t] = pim[t];
  }
  __syncthreads();

  const int p = threadIdx.x;
  if (p < nphi) {
    const float alpha = phi0 + (2.0f * PI) * (float)p / (float)nphi;
    float s1, c1;
    __sincosf(alpha, &s1, &c1);
    float cm = c1, sm = s1;                  // e^{i*1*alpha}
    float acc = lre[0];                      // m = 0 term (Im ignored, w=1)
    for (int mm = 1; mm < nm; ++mm) {
      const float w = (mm == nm - 1) ? 1.0f : 2.0f;
      acc += w * (lre[mm] * cm - lim[mm] * sm);
      const float cn = cm * c1 - sm * s1;    // rotate by alpha
      const float sn = sm * c1 + cm * s1;
      cm = cn;
      sm = sn;
    }
    out[(size_t)bc * NPIX + start + p] = acc / (float)nphi;
  }
}

// ---------------------------------------------------------------------------
extern "C" void kernel_launch(void* const* d_in, const int* in_sizes, int n_in,
                              void* d_out, int out_size, void* d_ws, size_t ws_size,
                              hipStream_t stream) {
  (void)in_sizes; (void)n_in; (void)out_size; (void)ws_size;

  const float* xr  = (const float*)d_in[0];   // [BC, LMAX, MMAX]
  const float* xi  = (const float*)d_in[1];   // [BC, LMAX, MMAX]
  const float* pct = (const float*)d_in[2];   // [MMAX, LMAX, NLAT]
  float* out = (float*)d_out;                 // [BC, NPIX]

  float* wre = (float*)d_ws;                          // [BC, NLAT, MMAX]
  float* wim = wre + (size_t)BC * NLAT * MMAX;        // [BC, NLAT, MMAX]

  dim3 gA(8, 8, MMAX);       // col pairs, bc row quads, m
  legendre_wmma_kernel<<<gA, dim3(128), 0, stream>>>(xr, xi, pct, wre, wim);

  dim3 gB(1, NLAT, BC);      // (pixels handled by 256 threads), ring, bc
  healpix_irfft_kernel<<<gB, dim3(256), 0, stream>>>(wre, wim, out);
}